// KernelMetricLoss_20426864460163
// MI455X (gfx1250) — compile-verified
//
#include <hip/hip_runtime.h>
#include <hip/hip_bf16.h>
#include <math.h>

#define NCLS   32
#define DIM    64
#define GAMMA_ 1.0f
#define MARGIN_ 5.0f
#define ALPHA_ 0.5f
#define EPSV   1e-9f
#define MAXTILES 64   // supports up to 1024 members per class

typedef float v2f __attribute__((ext_vector_type(2)));
typedef float v8f __attribute__((ext_vector_type(8)));

// ---------------- Pass 1: per-point squared norms + class counts ----------------
__global__ void k_norms_counts(const float* __restrict__ F, const int* __restrict__ lab,
                               float* __restrict__ norms, int* __restrict__ counts, int N) {
    int i = blockIdx.x * blockDim.x + threadIdx.x;
    if (i >= N) return;
    const float4* r4 = reinterpret_cast<const float4*>(F + (size_t)i * DIM);
    float s = 0.f;
#pragma unroll
    for (int q = 0; q < DIM / 4; ++q) {
        float4 v = r4[q];
        s += v.x * v.x + v.y * v.y + v.z * v.z + v.w * v.w;
    }
    norms[i] = s;
    atomicAdd(&counts[lab[i]], 1);
}

// ---------------- Pass 2: exclusive scan over 32 class counts ----------------
__global__ void k_scan(const int* __restrict__ counts, int* __restrict__ offsets) {
    if (threadIdx.x == 0) {
        int acc = 0;
        for (int c = 0; c < NCLS; ++c) { offsets[c] = acc; acc += counts[c]; }
    }
}

// ---------------- Pass 3: scatter indices into class-sorted order ----------------
__global__ void k_scatter(const int* __restrict__ lab, const int* __restrict__ offsets,
                          int* __restrict__ fill, int* __restrict__ sorted, int N) {
    int i = blockIdx.x * blockDim.x + threadIdx.x;
    if (i >= N) return;
    int l = lab[i];
    int pos = offsets[l] + atomicAdd(&fill[l], 1);
    sorted[pos] = i;
}

// ---------------- Pass 4: per-class 16x16 Gram tiles via V_WMMA_F32_16X16X4_F32 ----------------
// One wave (32 threads) per (class, tileI, tileJ).
__global__ __launch_bounds__(32)
void k_intra(const float* __restrict__ F, const float* __restrict__ norms,
             const int* __restrict__ counts, const int* __restrict__ offsets,
             const int* __restrict__ sorted, float* __restrict__ clsSum) {
    const int c = blockIdx.z;
    const int m = counts[c];
    const int t = (m + 15) >> 4;
    if ((int)blockIdx.x >= t || (int)blockIdx.y >= t) return;

    const int base = offsets[c];
    const int lane = threadIdx.x;
    const int r  = lane & 15;   // A-row / B-col carried by this lane
    const int hi = lane >> 4;   // which K half-pair this lane supplies

    // tile-local rows for A side and B side
    const int iaT = (int)blockIdx.x * 16 + r;
    const int ibT = (int)blockIdx.y * 16 + r;
    const bool va = iaT < m;
    const bool vb = ibT < m;
    const int gA = sorted[base + (va ? iaT : 0)];
    const int gB = sorted[base + (vb ? ibT : 0)];

    const float* rowA = F + (size_t)gA * DIM;
    const float* rowB = F + (size_t)gB * DIM;

    // Share row metadata through LDS (each half-wave covers all 16 rows of one side).
    __shared__ int   sIdxA[16], sIdxB[16];
    __shared__ float sNA[16], sNB[16];
    if (lane < 16) { sIdxA[r] = va ? gA : -1; sNA[r] = norms[gA]; }
    else           { sIdxB[r] = vb ? gB : -1; sNB[r] = norms[gB]; }
    __syncthreads();

    // Gram accumulation: 16 chained f32 WMMAs over K = 64.
    v8f acc = {};
#pragma unroll
    for (int kk = 0; kk < DIM / 4; ++kk) {
        const int k = kk * 4 + hi * 2;
        v2f a = *reinterpret_cast<const v2f*>(rowA + k);
        v2f b = *reinterpret_cast<const v2f*>(rowB + k);
        acc = __builtin_amdgcn_wmma_f32_16x16x4_f32(
            /*neg_a=*/false, a, /*neg_b=*/false, b,
            /*c_mod=*/(short)0, acc, /*reuse_a=*/false, /*reuse_b=*/false);
    }

    // C layout: VGPR e -> M = e + hi*8 (lanes 0-15 / 16-31), N = lane&15.
    float sum = 0.f;
#pragma unroll
    for (int e = 0; e < 8; ++e) {
        const int M = hi * 8 + e;
        const int gi = sIdxA[M];
        const int gj = sIdxB[r];
        if (gi >= 0 && gj >= 0 && gi != gj) {
            float sq = sNA[M] + sNB[r] - 2.0f * acc[e];
            sq = fmaxf(sq, 0.0f);
            float K = expf(-GAMMA_ * sq);
            sum += sqrtf(2.0f * (1.0f - K + EPSV));
        }
    }
    // wave32 reduction
#pragma unroll
    for (int off = 16; off > 0; off >>= 1) sum += __shfl_xor(sum, off, 32);
    if (lane == 0) atomicAdd(&clsSum[c], sum);
}

// ---------------- Pass 5: deterministic per-class center sums ----------------
__global__ __launch_bounds__(64)
void k_centers(const float* __restrict__ F, const int* __restrict__ counts,
               const int* __restrict__ offsets, const int* __restrict__ sorted,
               float* __restrict__ centerSum) {
    const int c = blockIdx.x;
    const int k = threadIdx.x;
    const int m = counts[c];
    const int base = offsets[c];
    float s = 0.f;
    for (int i = 0; i < m; ++i) s += F[(size_t)sorted[base + i] * DIM + k];
    centerSum[c * DIM + k] = s;
}

// ---------------- Pass 6: inter-class hinge + final scalar ----------------
__global__ __launch_bounds__(1024)
void k_final(const float* __restrict__ centerSum, const int* __restrict__ counts,
             const float* __restrict__ clsSum, float* __restrict__ out) {
    __shared__ float sC[NCLS * DIM];
    __shared__ float sN[NCLS];
    __shared__ int   sCnt[NCLS];
    __shared__ float sHinge;

    const int tid = threadIdx.x;
    if (tid < NCLS) sCnt[tid] = counts[tid];
    if (tid == 0) sHinge = 0.f;
    __syncthreads();
    for (int idx = tid; idx < NCLS * DIM; idx += blockDim.x) {
        int c = idx >> 6;
        sC[idx] = centerSum[idx] / fmaxf((float)sCnt[c], 1.0f);
    }
    __syncthreads();
    if (tid < NCLS) {
        float s = 0.f;
        for (int k = 0; k < DIM; ++k) { float v = sC[tid * DIM + k]; s += v * v; }
        sN[tid] = s;
    }
    __syncthreads();

    const int c1 = tid >> 5;
    const int c2 = tid & 31;
    bool valid = (sCnt[c1] > 0) && (sCnt[c2] > 0) && (c1 != c2);
    if (valid) {
        float dot = 0.f;
        for (int k = 0; k < DIM; ++k) dot += sC[c1 * DIM + k] * sC[c2 * DIM + k];
        float sq = fmaxf(sN[c1] + sN[c2] - 2.0f * dot, 0.0f);
        float K = expf(-GAMMA_ * sq);
        float CD = sqrtf(2.0f * (1.0f - K + EPSV));
        float h = fmaxf(MARGIN_ - CD, 0.0f);
        atomicAdd(&sHinge, h);
    }
    __syncthreads();

    if (tid == 0) {
        int P = 0;
        for (int c = 0; c < NCLS; ++c) P += (sCnt[c] > 0);
        float numU = fmaxf((float)P, 1.0f);
        float intra = 0.f;
        for (int c = 0; c < NCLS; ++c) {
            float cnt = (float)sCnt[c];
            if (sCnt[c] >= 2) intra += clsSum[c] / (cnt * (cnt - 1.0f));
        }
        intra /= numU;
        int nvalid = P * (P - 1);
        float inter = (P >= 2) ? (sHinge / fmaxf((float)nvalid, 1.0f)) : 0.0f;
        out[0] = intra + ALPHA_ * inter;
    }
}

extern "C" void kernel_launch(void* const* d_in, const int* in_sizes, int n_in,
                              void* d_out, int out_size, void* d_ws, size_t ws_size,
                              hipStream_t stream) {
    const float* F   = (const float*)d_in[0];
    const int*   lab = (const int*)d_in[1];
    const int N = in_sizes[1];           // 8192
    float* out = (float*)d_out;

    // workspace layout
    char* ws = (char*)d_ws;
    int*   counts    = (int*)(ws + 0);          // 32 ints
    int*   fill      = (int*)(ws + 128);        // 32 ints
    int*   offsets   = (int*)(ws + 256);        // 32 ints
    float* clsSum    = (float*)(ws + 384);      // 32 floats
    float* centerSum = (float*)(ws + 512);      // 32*64 floats = 8192 B
    float* norms     = (float*)(ws + 512 + NCLS * DIM * sizeof(float));
    int*   sorted    = (int*)((char*)norms + (size_t)N * sizeof(float));

    // zero counts/fill/offsets/clsSum (centerSum fully overwritten later)
    hipMemsetAsync(d_ws, 0, 512, stream);

    const int TB = 256;
    k_norms_counts<<<(N + TB - 1) / TB, TB, 0, stream>>>(F, lab, norms, counts, N);
    k_scan<<<1, 32, 0, stream>>>(counts, offsets);
    k_scatter<<<(N + TB - 1) / TB, TB, 0, stream>>>(lab, offsets, fill, sorted, N);
    k_intra<<<dim3(MAXTILES, MAXTILES, NCLS), 32, 0, stream>>>(F, norms, counts, offsets,
                                                               sorted, clsSum);
    k_centers<<<NCLS, DIM, 0, stream>>>(F, counts, offsets, sorted, centerSum);
    k_final<<<1, 1024, 0, stream>>>(centerSum, counts, clsSum, out);
}